// GraphConv_72078141161881
// MI455X (gfx1250) — compile-verified
//
#include <hip/hip_runtime.h>

// GraphConv for MI455X (gfx1250, wave32):
//   out[:, 0:64]   = features @ W              (f32 WMMA 16x16x4)
//   out[:, 64:128] = segment_mean(features[src], dst) @ W
//
// Phase 1: zero workspace (counts + summed accumulators)
// Phase 2: edge scatter with global f32 atomics (L2-resident: ~31 MB working set)
// Phase 3: fused dual GEMM with V_WMMA_F32_16X16X4_F32, Wt staged in LDS

#define N_NODES 50000
#define N_EDGES 800000
#define FEAT    64
#define CNT_PAD 50048            // counts region padded to multiple of 4 floats

typedef __attribute__((ext_vector_type(2))) float v2f;
typedef __attribute__((ext_vector_type(8))) float v8f;

__global__ void zero_ws_kernel(float4* __restrict__ ws, int n4) {
    int stride = gridDim.x * blockDim.x;
    float4 z = make_float4(0.f, 0.f, 0.f, 0.f);
    for (int i = blockIdx.x * blockDim.x + threadIdx.x; i < n4; i += stride)
        ws[i] = z;
}

__global__ void scatter_kernel(const float* __restrict__ feat,
                               const int*   __restrict__ edge_dst,
                               const int*   __restrict__ edge_src,
                               float*       __restrict__ counts,
                               float*       __restrict__ summed) {
    const long total  = (long)N_EDGES * FEAT;
    const long stride = (long)gridDim.x * blockDim.x;
    for (long idx = (long)blockIdx.x * blockDim.x + threadIdx.x; idx < total; idx += stride) {
        int e = (int)(idx >> 6);
        int f = (int)(idx & 63);
        // Hide index-load latency for the next grid-stride iteration
        // (speculative global_prefetch_b8; OOB translation failures are dropped).
        long nidx = idx + stride;
        if (nidx < total) {
            int ne = (int)(nidx >> 6);
            __builtin_prefetch(&edge_dst[ne], 0, 0);
            __builtin_prefetch(&edge_src[ne], 0, 0);
        }
        int d = edge_dst[e];
        int s = edge_src[e];
        atomicAdd(&summed[(long)d * FEAT + f], feat[(long)s * FEAT + f]);
        if (f == 0) atomicAdd(&counts[d], 1.0f);
    }
}

// One wave produces a 16-row x 64-col tile of BOTH gemms (nodes_rep and agg).
// 8 waves / block -> 128 rows per block.
__global__ void __launch_bounds__(256) gconv_wmma_kernel(
        const float* __restrict__ feat,      // [N, 64]
        const float* __restrict__ weight,    // [64 in, 64 out] row-major
        const float* __restrict__ counts,    // [N]
        const float* __restrict__ summed,    // [N, 64]
        float*       __restrict__ out)       // [N, 128]
{
    __shared__ float WT[FEAT * FEAT];        // WT[n*64+k] = W[k*64+n] (transposed)
    for (int i = threadIdx.x; i < FEAT * FEAT; i += blockDim.x) {
        int k = i >> 6, n = i & 63;
        WT[n * FEAT + k] = weight[i];
    }
    __syncthreads();

    const int wave = threadIdx.x >> 5;       // wave32
    const int lane = threadIdx.x & 31;
    const int r    = lane & 15;              // position within 16-lane half
    const int hi   = lane >> 4;              // 0: lanes 0-15, 1: lanes 16-31

    const int tile = blockIdx.x * 8 + wave;  // 16-row tile index
    if (tile >= N_NODES / 16) return;        // uniform per-wave: EXEC stays all-1s
    const int row0 = tile * 16;
    const int row  = row0 + r;               // this lane's A-fragment row

    const float rcp = 1.0f / fmaxf(counts[row], 1.0f);

    v8f acc1[4] = {};                        // features @ W, n-tiles 0..3
    v8f acc2[4] = {};                        // mean     @ W, n-tiles 0..3

    const float* frow = feat   + (long)row * FEAT;
    const float* srow = summed + (long)row * FEAT;

#pragma unroll
    for (int k = 0; k < FEAT; k += 4) {
        // A fragment: lanes 0-15 hold K=k,k+1 ; lanes 16-31 hold K=k+2,k+3
        const int kk = k + hi * 2;
        v2f a1 = *(const v2f*)(frow + kk);
        v2f s  = *(const v2f*)(srow + kk);
        v2f a2; a2.x = s.x * rcp; a2.y = s.y * rcp;
#pragma unroll
        for (int t = 0; t < 4; ++t) {
            const int n = t * 16 + r;        // B: N striped across 16-lane halves
            v2f b = *(const v2f*)(&WT[n * FEAT + kk]);   // W[kk..kk+1][n]
            acc1[t] = __builtin_amdgcn_wmma_f32_16x16x4_f32(
                          false, a1, false, b, (short)0, acc1[t], false, false);
            acc2[t] = __builtin_amdgcn_wmma_f32_16x16x4_f32(
                          false, a2, false, b, (short)0, acc2[t], false, false);
        }
    }

    // C/D layout: VGPR v -> M = v + hi*8, N = r.  out row stride = 128.
#pragma unroll
    for (int v = 0; v < 8; ++v) {
        const long orow = (long)(row0 + v + hi * 8) * 128;
#pragma unroll
        for (int t = 0; t < 4; ++t) {
            out[orow +      t * 16 + r] = acc1[t][v];
            out[orow + 64 + t * 16 + r] = acc2[t][v];
        }
    }
}

extern "C" void kernel_launch(void* const* d_in, const int* in_sizes, int n_in,
                              void* d_out, int out_size, void* d_ws, size_t ws_size,
                              hipStream_t stream) {
    (void)in_sizes; (void)n_in; (void)out_size; (void)ws_size;
    const float* feat     = (const float*)d_in[0];
    const float* weight   = (const float*)d_in[1];
    const int*   edge_dst = (const int*)d_in[2];
    const int*   edge_src = (const int*)d_in[3];
    float*       out      = (float*)d_out;

    float* counts = (float*)d_ws;            // [CNT_PAD]
    float* summed = counts + CNT_PAD;        // [N_NODES * FEAT]

    // Phase 1: zero accumulators (~13 MB)
    const int totalf = CNT_PAD + N_NODES * FEAT;
    const int n4     = totalf / 4;
    int zb = (n4 + 255) / 256;
    zero_ws_kernel<<<zb, 256, 0, stream>>>((float4*)d_ws, n4);

    // Phase 2: edge scatter (atomic segment-sum + counts)
    scatter_kernel<<<4096, 256, 0, stream>>>(feat, edge_dst, edge_src, counts, summed);

    // Phase 3: fused dual GEMM via f32 WMMA. 3125 row-tiles, 8 waves/block.
    const int tiles  = N_NODES / 16;         // 3125 exactly
    const int blocks = (tiles + 7) / 8;      // 391
    gconv_wmma_kernel<<<blocks, 256, 0, stream>>>(feat, weight, counts, summed, out);
}